// TDMMPostModel_33990371180742
// MI455X (gfx1250) — compile-verified
//
#include <hip/hip_runtime.h>
#include <math.h>

// ---------------- problem constants ----------------
#define BB     16
#define HH     160
#define WW     160
#define HW     25600         // 160*160
#define TOPK   32
#define NC     91
#define KDIM   80            // 79 params padded to 80 (multiple of 4 for WMMA K)
#define NV     204
#define NPAD   208           // 204 padded to 13*16
#define MTOT   512           // B*TOPK
#define RAD2DEG 57.29577951308232f
#define INF_F  __builtin_inff()

typedef __attribute__((ext_vector_type(2))) float v2f;
typedef __attribute__((ext_vector_type(8))) float v8f;

// ---------------- workspace layout (float indices) ----------------
#define WS_PEAK   0          // 409600  : peak scores per pixel
#define WS_SCORE  409600     // 512     : top-k scores
#define WS_IDX    410112     // 512     : top-k flat pixel idx (int)
#define WS_A      410624     // 512*80  : GEMM A operand (row-major, K=80)
#define WS_S      451584     // 512     : scale s
#define WS_R      452096     // 512*9   : rotation R row-major
#define WS_BK     456704     // 80*208  : GEMM B operand, Bk[k][v] (k-major)
#define WS_VERTS  473344     // 512*208 : GEMM result + u_base
#define WS_LNM    579840     // 512*136 : landmarks yx
#define WS_BBOX   649472     // 512*5   : candidate bboxes [y1,x1,y2,x2,score]
#define WS_POSE   652032     // 512*3   : pitch,yaw,roll
// total 653568 floats = ~2.5 MB

// ============================================================
// K1a: 3x3 SAME max-pool peak detection. peak = (v==nbhd max) ? v : 0
// ============================================================
__global__ void k_peaks(const float* __restrict__ hms, float* __restrict__ peak) {
    int p = blockIdx.x * blockDim.x + threadIdx.x;        // 409600 threads
    int b = p / HW;
    int rem = p - b * HW;
    int y = rem / WW;
    int x = rem - y * WW;
    float v = hms[p];
    float mx = v;
    int y0 = (y > 0) ? y - 1 : y,   y1 = (y < HH - 1) ? y + 1 : y;
    int x0 = (x > 0) ? x - 1 : x,   x1 = (x < WW - 1) ? x + 1 : x;
    for (int yy = y0; yy <= y1; ++yy)
        for (int xx = x0; xx <= x1; ++xx)
            mx = fmaxf(mx, hms[b * HW + yy * WW + xx]);
    peak[p] = (mx == v) ? v : 0.0f;
}

// ============================================================
// K1b: per-batch iterative top-32 argmax with lexicographic key
// (value desc, index asc). No buffer mutation: each round selects the
// max element whose key is strictly below the previous winner's key.
// ============================================================
__global__ void k_topk(const float* __restrict__ peak,
                       float* __restrict__ oscore, int* __restrict__ oidx) {
    __shared__ float sv[256];
    __shared__ int   si[256];
    __shared__ float s_pv;
    __shared__ int   s_pi;
    int b = blockIdx.x, tid = threadIdx.x;
    const float* pk = peak + b * HW;
    if (tid == 0) { s_pv = INF_F; s_pi = -1; }
    __syncthreads();
    for (int it = 0; it < TOPK; ++it) {
        float pv = s_pv; int pi = s_pi;
        float best = -INF_F; int bidx = HW;
        for (int i = tid; i < HW; i += 256) {
            float v = pk[i];
            // strictly below previous key (v,-i)
            bool ok = (v < pv) || (v == pv && i > pi);
            if (ok && (v > best || (v == best && i < bidx))) { best = v; bidx = i; }
        }
        sv[tid] = best; si[tid] = bidx;
        __syncthreads();
        for (int s = 128; s > 0; s >>= 1) {
            if (tid < s) {
                float v2 = sv[tid + s]; int i2 = si[tid + s];
                if (v2 > sv[tid] || (v2 == sv[tid] && i2 < si[tid])) { sv[tid] = v2; si[tid] = i2; }
            }
            __syncthreads();
        }
        if (tid == 0) {
            oscore[b * TOPK + it] = sv[0];
            oidx  [b * TOPK + it] = si[0];
            s_pv = sv[0]; s_pi = si[0];
        }
        __syncthreads();
    }
}

// ============================================================
// K2: gather 91 params at each peak, descale, split into s / R / A-row
// ============================================================
__global__ void k_gather(const float* __restrict__ pmaps, const float* __restrict__ pms,
                         const int* __restrict__ idx,
                         float* __restrict__ A, float* __restrict__ sv, float* __restrict__ Rw) {
    int m = blockIdx.x * blockDim.x + threadIdx.x;        // 512 threads
    if (m >= MTOT) return;
    int b = m >> 5;
    int id = idx[m];
    int y = id / WW, x = id - y * WW;
    const float* src = pmaps + ((long)(b * HH + y) * WW + x) * NC;
    const float* mean = pms;          // pms[0][:]
    const float* stdv = pms + NC;     // pms[1][:]
    float rt[11];
    for (int c = 0; c < NC; ++c) {
        float pv = src[c] * stdv[c] + mean[c];
        if (c == 0)            sv[m] = pv;
        else if (c < 12)       rt[c - 1] = pv;
        else                   A[m * KDIM + (c - 12)] = pv;   // shp(50) then exp(29) -> k 0..78
    }
    A[m * KDIM + 79] = 0.0f;
    // R[r][c] = rt12[4r+c], c<3  (rt12 = [rt0..rt10, 1])
    float* R = Rw + m * 9;
    R[0] = rt[0]; R[1] = rt[1]; R[2] = rt[2];
    R[3] = rt[4]; R[4] = rt[5]; R[5] = rt[6];
    R[6] = rt[8]; R[7] = rt[9]; R[8] = rt[10];
}

// ============================================================
// K2b: build B operand Bk[k][v] = concat(shp_base,exp_base)^T, zero padded
// ============================================================
__global__ void k_bmat(const float* __restrict__ shp_base, const float* __restrict__ exp_base,
                       float* __restrict__ Bk) {
    int t = blockIdx.x * blockDim.x + threadIdx.x;        // 80*208 = 16640 threads
    int k = t / NPAD, v = t - k * NPAD;
    float val = 0.0f;
    if (v < NV) {
        if (k < 50)      val = shp_base[v * 50 + k];
        else if (k < 79) val = exp_base[v * 29 + (k - 50)];
    }
    Bk[t] = val;
}

// ============================================================
// K3: verts = A(512x80) * Bk(80x208) via V_WMMA_F32_16X16X4_F32, + u_base
// one wave per 16x16 tile; 32 M-tiles x 13 N-tiles = 416 waves exactly.
// A frag  : lanes 0-15 hold M=0..15 {K=k0,k0+1}; lanes 16-31 hold {K=k0+2,k0+3}
// B frag  : mirrored half-split across lanes (V0/V1 = K even/odd rows)
// C/D frag: vgpr g, lane l -> (M = g + 8*(l>>4), N = l&15)
// ============================================================
__global__ __launch_bounds__(128) void k_gemm(const float* __restrict__ A,
                                              const float* __restrict__ Bk,
                                              const float* __restrict__ u_base,
                                              float* __restrict__ verts) {
    int lane = threadIdx.x & 31;
    int wid  = threadIdx.x >> 5;
    int gw   = blockIdx.x * 4 + wid;       // 0..415, no remainder waves
    int mt   = gw / 13;
    int nt   = gw - mt * 13;
    int l15  = lane & 15;
    int hi   = lane >> 4;
    int mrow = mt * 16 + l15;
    int khi  = hi * 2;
    int vcol = nt * 16 + l15;

    v8f acc = {};
    for (int kb = 0; kb < KDIM; kb += 4) {
        v2f a, bf;
        a.x  = A[mrow * KDIM + kb + khi];
        a.y  = A[mrow * KDIM + kb + khi + 1];
        bf.x = Bk[(kb + khi)     * NPAD + vcol];
        bf.y = Bk[(kb + khi + 1) * NPAD + vcol];
        acc = __builtin_amdgcn_wmma_f32_16x16x4_f32(
            /*neg_a=*/false, a, /*neg_b=*/false, bf,
            /*c_mod=*/(short)0, acc, /*reuse_a=*/false, /*reuse_b=*/false);
    }
    float u = (vcol < NV) ? u_base[vcol] : 0.0f;
#pragma unroll
    for (int g = 0; g < 8; ++g) {
        int m = mt * 16 + g + 8 * hi;
        verts[m * NPAD + vcol] = acc[g] + u;
    }
}

// ============================================================
// K4: landmarks (yx), candidate bbox, pose per candidate
// ============================================================
__global__ void k_lnmks(const float* __restrict__ verts, const float* __restrict__ sv,
                        const float* __restrict__ Rw, const float* __restrict__ score,
                        const int* __restrict__ idx, const float* __restrict__ oshapes,
                        float* __restrict__ lnm, float* __restrict__ bbox,
                        float* __restrict__ pose) {
    int m = blockIdx.x * blockDim.x + threadIdx.x;        // 512 threads
    if (m >= MTOT) return;
    int b = m >> 5;
    float s = sv[m];
    float R[9];
    for (int i = 0; i < 9; ++i) R[i] = Rw[m * 9 + i];
    int id = idx[m];
    float ys = (float)(id / WW), xs = (float)(id - (id / WW) * WW);
    float r0 = oshapes[b * 2 + 0] / 160.0f;
    float r1 = oshapes[b * 2 + 1] / 160.0f;
    float offy = ys * r0, offx = xs * r1;
    const float* V = verts + m * NPAD;
    float miny = INF_F, minx = INF_F, maxy = -INF_F, maxx = -INF_F;
    for (int v = 0; v < 68; ++v) {
        float v0 = V[3 * v], v1 = V[3 * v + 1], v2 = V[3 * v + 2];
        float px = s * (v0 * R[0] + v1 * R[1] + v2 * R[2]);   // r=0 component
        float py = s * (v0 * R[3] + v1 * R[4] + v2 * R[5]);   // r=1 component
        float ly = py + offy;
        float lx = px + offx;
        lnm[m * 136 + 2 * v]     = ly;
        lnm[m * 136 + 2 * v + 1] = lx;
        miny = fminf(miny, ly); maxy = fmaxf(maxy, ly);
        minx = fminf(minx, lx); maxx = fmaxf(maxx, lx);
    }
    float sc = score[m];
    bool mask = sc > 0.5f;
    bbox[m * 5 + 0] = mask ? miny : -1.0f;
    bbox[m * 5 + 1] = mask ? minx : -1.0f;
    bbox[m * 5 + 2] = mask ? maxy : -1.0f;
    bbox[m * 5 + 3] = mask ? maxx : -1.0f;
    bbox[m * 5 + 4] = mask ? sc   : -1.0f;
    // pose (note: reference takes cos() of yaw in DEGREES — replicate exactly)
    float yaw = asinf(-R[6]) * RAD2DEG;
    float cy = cosf(yaw);
    float pitch = atan2f(R[7] / cy, R[8] / cy) * RAD2DEG;
    float roll  = atan2f(R[3] / cy, R[0] / cy) * RAD2DEG;
    pose[m * 3 + 0] = pitch;
    pose[m * 3 + 1] = yaw;
    pose[m * 3 + 2] = roll;
}

// ============================================================
// K5: NMS (stable sort by score desc, sequential suppression), scatter by
// rank, sentinel plumbing (-1 -> inf, 0 -> inf), matched mask, final writes.
// One wave (32 lanes) per batch.
// ============================================================
__global__ __launch_bounds__(32) void k_nms(const float* __restrict__ bbox,
                                            const float* __restrict__ score,
                                            const float* __restrict__ pose,
                                            const float* __restrict__ lnm,
                                            float* __restrict__ out) {
    __shared__ float sbox[32][4], ssc[32];
    __shared__ int   pos[32];
    __shared__ float sortb[32][4], sorts[32];
    __shared__ int   ssupp[32], karr[32];
    __shared__ float ob[32][4], osc[32];
    __shared__ int   dv[32];
    int b = blockIdx.x, j = threadIdx.x;
    const float* bb = bbox + (b * 32 + j) * 5;
    float b0 = bb[0], b1 = bb[1], b2 = bb[2], b3 = bb[3], sc = bb[4];
    sbox[j][0] = b0; sbox[j][1] = b1; sbox[j][2] = b2; sbox[j][3] = b3; ssc[j] = sc;
    __syncthreads();
    // stable descending rank
    int rank = 0;
    for (int i = 0; i < 32; ++i) {
        float si = ssc[i];
        rank += (si > sc) || (si == sc && i < j);
    }
    pos[rank] = j;
    __syncthreads();
    int src = pos[j];
    float bs0 = sbox[src][0], bs1 = sbox[src][1], bs2 = sbox[src][2], bs3 = sbox[src][3];
    float ss = ssc[src];
    sortb[j][0] = bs0; sortb[j][1] = bs1; sortb[j][2] = bs2; sortb[j][3] = bs3; sorts[j] = ss;
    float areaj = (bs2 - bs0) * (bs3 - bs1);
    ssupp[j] = 0;
    __syncthreads();
    // sequential suppression
    for (int i = 0; i < 32; ++i) {
        int supi = ssupp[i];
        float i0 = sortb[i][0], i1 = sortb[i][1], i2 = sortb[i][2], i3 = sortb[i][3];
        float ai = (i2 - i0) * (i3 - i1);
        float yy1 = fmaxf(i0, bs0), xx1 = fmaxf(i1, bs1);
        float yy2 = fminf(i2, bs2), xx2 = fminf(i3, bs3);
        float inter = fmaxf(yy2 - yy1, 0.0f) * fmaxf(xx2 - xx1, 0.0f);
        float uni = ai + areaj - inter;
        float iou = inter / fmaxf(uni, 1e-8f);
        if (!supi && j > i && iou > 0.4f) ssupp[j] = 1;
        __syncthreads();
    }
    int keep = !ssupp[j];
    karr[j] = keep;
    ob[j][0] = 0.0f; ob[j][1] = 0.0f; ob[j][2] = 0.0f; ob[j][3] = 0.0f; osc[j] = 0.0f;
    __syncthreads();
    int rank2 = 0;
    for (int i = 0; i < j; ++i) rank2 += karr[i];
    if (keep && rank2 < 32) {
        ob[rank2][0] = bs0; ob[rank2][1] = bs1; ob[rank2][2] = bs2; ob[rank2][3] = bs3;
        osc[rank2] = ss;
    }
    __syncthreads();
    // sentinel plumbing + det_valid + bbox output
    float o[4];
    bool fin = true;
    for (int c = 0; c < 4; ++c) {
        float v = ob[j][c];
        v = (v == -1.0f || v == 0.0f) ? INF_F : v;
        o[c] = v;
        fin = fin && (v != INF_F);
    }
    dv[j] = fin ? 1 : 0;
    float* ob_out = out + (b * 32 + j) * 6;
    ob_out[0] = o[0]; ob_out[1] = o[1]; ob_out[2] = o[2]; ob_out[3] = o[3];
    ob_out[4] = osc[j]; ob_out[5] = 0.0f;
    float myscore = score[b * 32 + j];
    __syncthreads();
    // matched: candidate's raw top-k score equals a valid detection's score
    int matched = 0;
    for (int i = 0; i < 32; ++i)
        matched |= (myscore == osc[i]) && dv[i];
    // pose output (base 72704)
    for (int c = 0; c < 3; ++c) {
        float v = pose[(b * 32 + j) * 3 + c];
        v = matched ? v : -1.0f;
        v = (v == -1.0f) ? INF_F : v;
        out[72704 + (b * 32 + j) * 3 + c] = v;
    }
    // lnmks output (base 3072)
    for (int t = 0; t < 136; ++t) {
        float v = lnm[(b * 32 + j) * 136 + t];
        v = matched ? v : -1.0f;
        v = (v == -1.0f) ? INF_F : v;
        out[3072 + (b * 32 + j) * 136 + t] = v;
    }
}

// ============================================================
extern "C" void kernel_launch(void* const* d_in, const int* in_sizes, int n_in,
                              void* d_out, int out_size, void* d_ws, size_t ws_size,
                              hipStream_t stream) {
    const float* hms      = (const float*)d_in[0];
    const float* pmaps    = (const float*)d_in[1];
    const float* oshapes  = (const float*)d_in[2];
    const float* pms      = (const float*)d_in[3];
    const float* u_base   = (const float*)d_in[4];
    const float* shp_base = (const float*)d_in[5];
    const float* exp_base = (const float*)d_in[6];
    float* out = (float*)d_out;
    float* ws  = (float*)d_ws;

    k_peaks <<<1600, 256, 0, stream>>>(hms, ws + WS_PEAK);
    k_topk  <<<BB,   256, 0, stream>>>(ws + WS_PEAK, ws + WS_SCORE, (int*)(ws + WS_IDX));
    k_gather<<<2,    256, 0, stream>>>(pmaps, pms, (const int*)(ws + WS_IDX),
                                       ws + WS_A, ws + WS_S, ws + WS_R);
    k_bmat  <<<65,   256, 0, stream>>>(shp_base, exp_base, ws + WS_BK);
    k_gemm  <<<104,  128, 0, stream>>>(ws + WS_A, ws + WS_BK, u_base, ws + WS_VERTS);
    k_lnmks <<<2,    256, 0, stream>>>(ws + WS_VERTS, ws + WS_S, ws + WS_R,
                                       ws + WS_SCORE, (const int*)(ws + WS_IDX), oshapes,
                                       ws + WS_LNM, ws + WS_BBOX, ws + WS_POSE);
    k_nms   <<<BB,   32,  0, stream>>>(ws + WS_BBOX, ws + WS_SCORE, ws + WS_POSE,
                                       ws + WS_LNM, out);
}